// C4OpcodeExecutor_62380105007577
// MI455X (gfx1250) — compile-verified
//
#include <hip/hip_runtime.h>
#include <cstdint>

// ---------------------------------------------------------------------------
// C4OpcodeExecutor: read_val = LE-assemble 8 bytes at read_addr per row;
// new_mem = memory with value's 8 bytes scattered at write_addr per row.
// Bandwidth-bound: ~0.5 GiB in + ~0.5 GiB out -> ~46 us floor @ 23.3 TB/s.
// Bulk copy: CDNA5 async global->LDS DMA (NT hint), double-buffered,
// 8 x B128 per lane per stage (offset: immediate applies to BOTH LDS and
// global addresses, per CDNA5 ISA), NT 128-bit output stores,
// all-32-bit indexing, no runtime division.
// ---------------------------------------------------------------------------

typedef __attribute__((ext_vector_type(4))) int   i32x4;
typedef __attribute__((ext_vector_type(4))) float f32x4;

#define BULK_BLOCK   256                      // 8 wave32 waves
#define OPS_PER_TILE 8                        // 8 x b128 async loads per lane
#define CHUNK_INTS   (BULK_BLOCK * 4)         // 1024 ints per b128 sweep
#define TILE_INTS    (CHUNK_INTS * OPS_PER_TILE)   // 8192 ints = 32 KiB
#define TILE_BYTES   (TILE_INTS * 4)
#define BULK_GRID    4096

// One asm block = 8 async DMA loads; INST_OFFSET is added to both the LDS
// destination and the global source, so one VGPR pair addresses all eight.
// TH_LOAD_NT: read-once 512 MiB stream, don't thrash L2.
#define ISSUE_TILE(ldsaddr, goff, base)                                                     \
    asm volatile("global_load_async_to_lds_b128 %0, %1, %2 offset:0 th:TH_LOAD_NT\n\t"      \
                 "global_load_async_to_lds_b128 %0, %1, %2 offset:4096 th:TH_LOAD_NT\n\t"   \
                 "global_load_async_to_lds_b128 %0, %1, %2 offset:8192 th:TH_LOAD_NT\n\t"   \
                 "global_load_async_to_lds_b128 %0, %1, %2 offset:12288 th:TH_LOAD_NT\n\t"  \
                 "global_load_async_to_lds_b128 %0, %1, %2 offset:16384 th:TH_LOAD_NT\n\t"  \
                 "global_load_async_to_lds_b128 %0, %1, %2 offset:20480 th:TH_LOAD_NT\n\t"  \
                 "global_load_async_to_lds_b128 %0, %1, %2 offset:24576 th:TH_LOAD_NT\n\t"  \
                 "global_load_async_to_lds_b128 %0, %1, %2 offset:28672 th:TH_LOAD_NT"      \
                 :: "v"(ldsaddr), "v"(goff), "s"(base) : "memory")

__global__ __launch_bounds__(BULK_BLOCK)
void bulk_copy_convert(const int* __restrict__ in, float* __restrict__ out,
                       uint32_t tilesQ, uint32_t tilesR,
                       uint32_t numTiles, long long total)
{
    __shared__ __align__(16) int buf[2][TILE_INTS];

    const uint32_t laneB    = (uint32_t)threadIdx.x * 16u;  // 16 B per lane
    const uint32_t laneInts = (uint32_t)threadIdx.x * 4u;
    // Low 32 bits of a flat LDS pointer are the wave-relative LDS byte addr.
    uint32_t lds[2];
    lds[0] = (uint32_t)(uintptr_t)buf[0] + laneB;
    lds[1] = (uint32_t)(uintptr_t)buf[1] + laneB;

    const uint32_t strideT = gridDim.x;
    const uint32_t b       = blockIdx.x;
    // Round-robin tile ownership; quotient/remainder supplied by host.
    const uint32_t myTiles = tilesQ + ((b < tilesR) ? 1u : 0u);

    uint32_t tLoad = b;                       // global tile id of next load
    // Prologue: up to 2 tiles (16 async instructions) in flight per wave.
    if (myTiles > 0u) { ISSUE_TILE(lds[0], tLoad * TILE_BYTES + laneB, in); tLoad += strideT; }
    if (myTiles > 1u) { ISSUE_TILE(lds[1], tLoad * TILE_BYTES + laneB, in); tLoad += strideT; }

    uint32_t tUse = b;
    int sel = 0;
    for (uint32_t i = 0; i < myTiles; ++i) {
        // Async loads complete in order. Two tiles in flight -> allow the
        // newer tile's 8 ops to stay outstanding; last tile -> drain fully.
        if (i + 1u < myTiles) asm volatile("s_wait_asynccnt 0x8" ::: "memory");
        else                  asm volatile("s_wait_asynccnt 0x0" ::: "memory");

        // Each lane consumes exactly the 8x16B its own async ops wrote.
        const uint32_t outBase = tUse * TILE_INTS + laneInts;
        #pragma unroll
        for (int k = 0; k < OPS_PER_TILE; ++k) {
            i32x4 d = *reinterpret_cast<const i32x4*>(
                          &buf[sel][laneInts + (uint32_t)k * CHUNK_INTS]);
            f32x4 f = __builtin_convertvector(d, f32x4);
            __builtin_nontemporal_store(
                f, reinterpret_cast<f32x4*>(out + outBase + (uint32_t)k * CHUNK_INTS));
        }

        // Async DMA LDS writes are unordered vs in-flight ds_loads: retire
        // our LDS reads before re-targeting this buffer.
        asm volatile("s_wait_dscnt 0x0" ::: "memory");

        if (i + 2u < myTiles) {
            ISSUE_TILE(lds[sel], tLoad * TILE_BYTES + laneB, in);
            tLoad += strideT;
        }
        tUse += strideT;
        sel ^= 1;
    }

    // Scalar tail (unused for 2^27 elements, kept for generality).
    for (long long idx = (long long)numTiles * TILE_INTS
                       + (long long)blockIdx.x * blockDim.x + threadIdx.x;
         idx < total;
         idx += (long long)gridDim.x * blockDim.x) {
        out[idx] = (float)in[idx];
    }
}

// One thread per row: assemble read_val from the ORIGINAL input, then patch
// the 8 scattered output elements with value's bytes (later i wins on clip).
__global__ __launch_bounds__(256)
void fixup_rows(const int* __restrict__ mem,
                const int* __restrict__ raddr,
                const int* __restrict__ waddr,
                const int* __restrict__ value,
                float* __restrict__ out_rv,
                float* __restrict__ out_mem,
                int B, int M)
{
    int b = blockIdx.x * blockDim.x + threadIdx.x;
    if (b >= B) return;

    const int* row = mem + (size_t)b * (size_t)M;

    long long ra = (long long)raddr[b];
    unsigned long long rv = 0ull;
    #pragma unroll
    for (int i = 0; i < 8; ++i) {
        long long a = ra + i;
        if (a > (long long)(M - 1)) a = M - 1;
        if (a < 0) a = 0;
        rv |= (unsigned long long)(unsigned int)row[a] << (8 * i);
    }
    out_rv[b] = (float)rv;

    unsigned long long v = (unsigned long long)(unsigned int)value[b];
    long long wa = (long long)waddr[b];
    float* orow = out_mem + (size_t)b * (size_t)M;
    #pragma unroll
    for (int i = 0; i < 8; ++i) {
        long long a = wa + i;
        if (a > (long long)(M - 1)) a = M - 1;
        if (a < 0) a = 0;
        orow[a] = (float)(unsigned int)((v >> (8 * i)) & 255ull);
    }
}

extern "C" void kernel_launch(void* const* d_in, const int* in_sizes, int n_in,
                              void* d_out, int out_size, void* d_ws, size_t ws_size,
                              hipStream_t stream)
{
    const int* mem   = (const int*)d_in[0];   // (B, M) byte values 0..255
    const int* raddr = (const int*)d_in[1];   // (B,)
    const int* waddr = (const int*)d_in[2];   // (B,)
    const int* value = (const int*)d_in[3];   // (B,)

    const int       B     = in_sizes[1];                 // 2048
    const long long total = (long long)in_sizes[0];      // B * M = 2^27
    const int       M     = (int)(total / (long long)B); // 65536

    float* out_rv  = (float*)d_out;          // first B floats: read_val
    float* out_mem = (float*)d_out + B;      // then B*M floats: new_mem

    const uint32_t numTiles = (uint32_t)(total / TILE_INTS);   // 16384
    const uint32_t tilesQ   = numTiles / BULK_GRID;            // 4
    const uint32_t tilesR   = numTiles % BULK_GRID;            // 0

    // 1) Bulk copy+convert via async global->LDS double-buffered DMA pipeline.
    bulk_copy_convert<<<BULK_GRID, BULK_BLOCK, 0, stream>>>(mem, out_mem,
                                                            tilesQ, tilesR,
                                                            numTiles, total);

    // 2) Ordered on the same stream: scattered byte writes + read_val gather.
    fixup_rows<<<(B + 255) / 256, 256, 0, stream>>>(mem, raddr, waddr, value,
                                                    out_rv, out_mem, B, M);
}